// GAT_39900246180202
// MI455X (gfx1250) — compile-verified
//
#include <hip/hip_runtime.h>

typedef __attribute__((ext_vector_type(2))) float v2f;
typedef __attribute__((ext_vector_type(8))) float v8f;

#define N_NODES 50000
#define E_EDGES 800000
#define E_TOT   (E_EDGES + N_NODES)
#define KDIM    256
#define NEG_SLOPE 0.2f

// ---------------------------------------------------------------------------
// Edge prep: int64 edge_index -> int32 src/dst with self-loops appended
// ---------------------------------------------------------------------------
__global__ void prep_edges(const long long* __restrict__ ei,
                           int* __restrict__ src, int* __restrict__ dst) {
  int t = blockIdx.x * blockDim.x + threadIdx.x;
  if (t >= E_TOT) return;
  if (t < E_EDGES) {
    src[t] = (int)ei[t];
    dst[t] = (int)ei[E_EDGES + t];
  } else {
    int n = t - E_EDGES;
    src[t] = n;
    dst[t] = n;
  }
}

// ---------------------------------------------------------------------------
// W [K x cols] -> Wt [cols x K]  (so WMMA B fragments load as contiguous f32x2)
// ---------------------------------------------------------------------------
__global__ void transpose_w(const float* __restrict__ W, float* __restrict__ Wt,
                            int cols) {
  int t = blockIdx.x * blockDim.x + threadIdx.x;
  if (t >= KDIM * cols) return;
  int k = t / cols, n = t % cols;
  Wt[n * KDIM + k] = W[k * cols + n];
}

__global__ void fill_zero_u32(unsigned* __restrict__ p, int n) {
  int t = blockIdx.x * blockDim.x + threadIdx.x;
  if (t < n) p[t] = 0u;
}

// ---------------------------------------------------------------------------
// f32 WMMA GEMM: Hout[rows x cols] = X[rows x 256] * W (given Wt[cols x 256]).
// One wave computes a 16x64 strip (4 col-tiles): A fragment loaded once per
// k-step feeds 4 x V_WMMA_F32_16X16X4_F32 (4x less A traffic from L2).
// A fragment (16x4 f32, 2 VGPR): lanes 0-15 {K=k+0,k+1}, lanes 16-31 {k+2,k+3}.
// ---------------------------------------------------------------------------
__global__ void gemm_wmma_f32(const float* __restrict__ X,
                              const float* __restrict__ Wt,
                              float* __restrict__ Hout, int rows, int cols) {
  int wave = (blockIdx.x * blockDim.x + threadIdx.x) >> 5;
  int lane = threadIdx.x & 31;
  int grp = cols >> 6;                         // 64-col groups: 4 (L0/L1), 1 (L2)
  int nWaves = (rows >> 4) * grp;
  if (wave >= nWaves) return;                  // wave-uniform: EXEC stays all-1s
  int row0 = (wave / grp) << 4;
  int col0 = (wave % grp) << 6;
  int m    = lane & 15;
  int koff = (lane >> 4) << 1;                 // 0 or 2
  const float* arow = X  + (row0 + m) * KDIM + koff;
  const float* brow = Wt + (col0 + m) * KDIM + koff;
  v8f a0 = {}, a1 = {}, a2 = {}, a3 = {};
  #pragma unroll 4
  for (int k = 0; k < KDIM; k += 4) {
    v2f a  = *(const v2f*)(arow + k);
    v2f b0 = *(const v2f*)(brow + k);
    v2f b1 = *(const v2f*)(brow + 16 * KDIM + k);
    v2f b2 = *(const v2f*)(brow + 32 * KDIM + k);
    v2f b3 = *(const v2f*)(brow + 48 * KDIM + k);
    a0 = __builtin_amdgcn_wmma_f32_16x16x4_f32(false, a, false, b0, (short)0, a0, false, false);
    a1 = __builtin_amdgcn_wmma_f32_16x16x4_f32(false, a, false, b1, (short)0, a1, false, false);
    a2 = __builtin_amdgcn_wmma_f32_16x16x4_f32(false, a, false, b2, (short)0, a2, false, false);
    a3 = __builtin_amdgcn_wmma_f32_16x16x4_f32(false, a, false, b3, (short)0, a3, false, false);
  }
  // D layout: VGPR i = D[row0 + (lane>=16?8:0) + i][tile_col0 + (lane&15)]
  int rowb = row0 + ((lane >> 4) << 3);
  int colw = col0 + (lane & 15);
  #pragma unroll
  for (int i = 0; i < 8; ++i) {
    float* o = Hout + (rowb + i) * cols + colw;
    o[0]  = a0[i];
    o[16] = a1[i];
    o[32] = a2[i];
    o[48] = a3[i];
  }
}

// ---------------------------------------------------------------------------
// a_src[n,h] = sum_c h[n,h,c]*att_src[h,c] ; a_dst likewise.
// One block per node, blockDim = heads*64; LDS tree-reduce per 64-group.
// ---------------------------------------------------------------------------
__global__ void attn_dots(const float* __restrict__ Hf,
                          const float* __restrict__ as,
                          const float* __restrict__ ad,
                          float* __restrict__ asrc, float* __restrict__ adst,
                          int heads) {
  int n = blockIdx.x;
  int t = threadIdx.x;                         // 0 .. heads*64-1
  int F = heads << 6;
  float hv = Hf[n * F + t];
  __shared__ float s1[256];
  __shared__ float s2[256];
  s1[t] = hv * as[t];
  s2[t] = hv * ad[t];
  __syncthreads();
  #pragma unroll
  for (int off = 32; off >= 1; off >>= 1) {
    if ((t & 63) < off) { s1[t] += s1[t + off]; s2[t] += s2[t + off]; }
    __syncthreads();
  }
  if ((t & 63) == 0) {
    int hh = t >> 6;
    asrc[n * heads + hh] = s1[t];
    adst[n * heads + hh] = s2[t];
  }
}

// ---------------------------------------------------------------------------
// e = leakyrelu(a_src[src] + a_dst[dst]); store; segment-max via ordered-uint
// ---------------------------------------------------------------------------
__device__ __forceinline__ unsigned f32_ord_enc(float v) {
  unsigned b = __float_as_uint(v);
  return b ^ ((b >> 31) ? 0xFFFFFFFFu : 0x80000000u);
}

__global__ void edge_score_max(const int* __restrict__ src,
                               const int* __restrict__ dst,
                               const float* __restrict__ asrc,
                               const float* __restrict__ adst,
                               float* __restrict__ eval,
                               unsigned* __restrict__ menc, int heads) {
  int e = blockIdx.x * blockDim.x + threadIdx.x;
  if (e >= E_TOT) return;
  int s = src[e], d = dst[e];
  for (int hh = 0; hh < heads; ++hh) {
    float v = asrc[s * heads + hh] + adst[d * heads + hh];
    v = v > 0.0f ? v : NEG_SLOPE * v;
    eval[e * heads + hh] = v;
    atomicMax(&menc[d * heads + hh], f32_ord_enc(v));
  }
}

// decode ordered-uint max back to float, in place
__global__ void decode_max(unsigned* __restrict__ m, int n) {
  int t = blockIdx.x * blockDim.x + threadIdx.x;
  if (t >= n) return;
  unsigned u = m[t];
  unsigned b = (u & 0x80000000u) ? (u ^ 0x80000000u) : ~u;
  ((float*)m)[t] = __uint_as_float(b);
}

// p = exp(e - m[dst]); store in place; denom[dst] += p
__global__ void edge_expsum(const int* __restrict__ dst,
                            float* __restrict__ ep,
                            const float* __restrict__ mf,
                            float* __restrict__ denom, int heads) {
  int e = blockIdx.x * blockDim.x + threadIdx.x;
  if (e >= E_TOT) return;
  int d = dst[e];
  for (int hh = 0; hh < heads; ++hh) {
    float p = __expf(ep[e * heads + hh] - mf[d * heads + hh]);
    ep[e * heads + hh] = p;
    atomicAdd(&denom[d * heads + hh], p);
  }
}

__global__ void recip_ip(float* __restrict__ d, int n) {
  int t = blockIdx.x * blockDim.x + threadIdx.x;
  if (t < n) d[t] = 1.0f / d[t];
}

// ---------------------------------------------------------------------------
// scatter (heads=4, F=256): 64 threads/edge, float4 chunks.
// Edge id is wave-uniform (wave32 spans half an edge's 64 chunks) ->
// readfirstlane lets per-edge scalars take the SMEM path.
// ---------------------------------------------------------------------------
__global__ void edge_scatter_h4(const int* __restrict__ src,
                                const int* __restrict__ dst,
                                const float* __restrict__ p,
                                const float* __restrict__ rden,
                                const float* __restrict__ Hf,
                                float* __restrict__ acc) {
  int t = blockIdx.x * blockDim.x + threadIdx.x;
  int e = __builtin_amdgcn_readfirstlane(t >> 6);
  if (e >= E_TOT) return;
  int c4 = t & 63;
  int hh = c4 >> 4;
  int cc = (c4 & 15) << 2;
  int s = src[e], d = dst[e];
  float alpha = p[e * 4 + hh] * rden[d * 4 + hh];
  const float4 hv = *(const float4*)(Hf + s * 256 + (hh << 6) + cc);
  float* o = acc + d * 256 + (hh << 6) + cc;
  atomicAdd(o + 0, hv.x * alpha);
  atomicAdd(o + 1, hv.y * alpha);
  atomicAdd(o + 2, hv.z * alpha);
  atomicAdd(o + 3, hv.w * alpha);
}

// scatter (heads=1, F=64): 32 threads/edge, float2 chunks; fully wave-uniform
// per-edge scalars (src/dst/p/rden all scalarizable).
__global__ void edge_scatter_h1(const int* __restrict__ src,
                                const int* __restrict__ dst,
                                const float* __restrict__ p,
                                const float* __restrict__ rden,
                                const float* __restrict__ Hf,
                                float* __restrict__ acc) {
  int t = blockIdx.x * blockDim.x + threadIdx.x;
  int e = __builtin_amdgcn_readfirstlane(t >> 5);
  if (e >= E_TOT) return;
  int cc = (t & 31) << 1;
  int s = src[e], d = dst[e];
  float alpha = p[e] * rden[d];
  const float2 hv = *(const float2*)(Hf + s * 64 + cc);
  float* o = acc + d * 64 + cc;
  atomicAdd(o + 0, hv.x * alpha);
  atomicAdd(o + 1, hv.y * alpha);
}

// out = (acc + bias) with optional leaky-relu
__global__ void bias_act(float* __restrict__ acc, const float* __restrict__ b,
                         int total, int fmask, int act) {
  int t = blockIdx.x * blockDim.x + threadIdx.x;
  if (t >= total) return;
  float v = acc[t] + b[t & fmask];
  if (act) v = v > 0.0f ? v : NEG_SLOPE * v;
  acc[t] = v;
}

// ---------------------------------------------------------------------------
extern "C" void kernel_launch(void* const* d_in, const int* in_sizes, int n_in,
                              void* d_out, int out_size, void* d_ws, size_t ws_size,
                              hipStream_t stream) {
  const float*      x  = (const float*)d_in[0];
  const long long*  ei = (const long long*)d_in[1];
  const float* W[3]  = {(const float*)d_in[2], (const float*)d_in[6], (const float*)d_in[10]};
  const float* AS[3] = {(const float*)d_in[3], (const float*)d_in[7], (const float*)d_in[11]};
  const float* AD[3] = {(const float*)d_in[4], (const float*)d_in[8], (const float*)d_in[12]};
  const float* B[3]  = {(const float*)d_in[5], (const float*)d_in[9], (const float*)d_in[13]};

  // ---- workspace carve-up (256B aligned) ----
  char* ws = (char*)d_ws;
  size_t off = 0;
  auto carve = [&](size_t bytes) -> char* {
    char* p = ws + off;
    off = (off + bytes + 255) & ~(size_t)255;
    return p;
  };
  int*      srcI = (int*)     carve((size_t)E_TOT * 4);
  int*      dstI = (int*)     carve((size_t)E_TOT * 4);
  float*    bufH = (float*)   carve((size_t)N_NODES * 256 * 4); // x@W
  float*    bufA = (float*)   carve((size_t)N_NODES * 256 * 4); // aggregate / next x
  float*    Wt   = (float*)   carve((size_t)256 * 256 * 4);
  float*    aS   = (float*)   carve((size_t)N_NODES * 4 * 4);
  float*    aD   = (float*)   carve((size_t)N_NODES * 4 * 4);
  unsigned* mB   = (unsigned*)carve((size_t)N_NODES * 4 * 4);   // enc-max -> float
  float*    dB   = (float*)   carve((size_t)N_NODES * 4 * 4);   // denom -> 1/denom
  float*    eP   = (float*)   carve((size_t)E_TOT * 4 * 4);     // e -> p
  (void)ws_size; (void)n_in; (void)in_sizes;

  const int TPB = 256;
  const int ebl = (E_TOT + TPB - 1) / TPB;

  prep_edges<<<ebl, TPB, 0, stream>>>(ei, srcI, dstI);

  // one GAT layer
  auto layer = [&](const float* xin, int heads, int cols, float* accOut,
                   const float* Wl, const float* asl, const float* adl,
                   const float* bl, int act) {
    int nh = N_NODES * heads;
    int tot = N_NODES * cols;
    // h = x @ W  (WMMA, 16x64 strip per wave)
    transpose_w<<<(KDIM * cols + TPB - 1) / TPB, TPB, 0, stream>>>(Wl, Wt, cols);
    int nWaves = (N_NODES >> 4) * (cols >> 6);
    gemm_wmma_f32<<<(nWaves * 32 + TPB - 1) / TPB, TPB, 0, stream>>>(
        xin, Wt, bufH, N_NODES, cols);
    // attention dots
    attn_dots<<<N_NODES, heads * 64, 0, stream>>>(bufH, asl, adl, aS, aD, heads);
    // segment softmax
    fill_zero_u32<<<(nh + TPB - 1) / TPB, TPB, 0, stream>>>(mB, nh);
    fill_zero_u32<<<(nh + TPB - 1) / TPB, TPB, 0, stream>>>((unsigned*)dB, nh);
    edge_score_max<<<ebl, TPB, 0, stream>>>(srcI, dstI, aS, aD, eP, mB, heads);
    decode_max<<<(nh + TPB - 1) / TPB, TPB, 0, stream>>>(mB, nh);
    edge_expsum<<<ebl, TPB, 0, stream>>>(dstI, eP, (const float*)mB, dB, heads);
    recip_ip<<<(nh + TPB - 1) / TPB, TPB, 0, stream>>>(dB, nh);
    // weighted scatter-add
    fill_zero_u32<<<(tot + TPB - 1) / TPB, TPB, 0, stream>>>((unsigned*)accOut, tot);
    if (heads == 4) {
      long long sth = (long long)E_TOT << 6;
      edge_scatter_h4<<<(int)((sth + TPB - 1) / TPB), TPB, 0, stream>>>(
          srcI, dstI, eP, dB, bufH, accOut);
    } else {
      long long sth = (long long)E_TOT << 5;
      edge_scatter_h1<<<(int)((sth + TPB - 1) / TPB), TPB, 0, stream>>>(
          srcI, dstI, eP, dB, bufH, accOut);
    }
    // bias (+ activation)
    bias_act<<<(tot + TPB - 1) / TPB, TPB, 0, stream>>>(accOut, bl, tot, cols - 1, act);
  };

  // layer 0: x[N,256] -> bufA[N,256], leaky-relu
  layer(x,    4, 256, bufA,          W[0], AS[0], AD[0], B[0], 1);
  // layer 1: bufA -> bufA (bufH scratch is free after its scatter), leaky-relu
  layer(bufA, 4, 256, bufA,          W[1], AS[1], AD[1], B[1], 1);
  // layer 2: bufA -> d_out[N,64], heads=1 (mean over 1 head = identity), no act
  layer(bufA, 1, 64,  (float*)d_out, W[2], AS[2], AD[2], B[2], 0);
  (void)out_size;
}